// GATNet_52879637348705
// MI455X (gfx1250) — compile-verified
//
#include <hip/hip_runtime.h>
#include <hip/hip_bf16.h>
#include <math.h>

typedef __attribute__((ext_vector_type(2))) float        v2f;
typedef __attribute__((ext_vector_type(8))) float        v8f;
typedef __attribute__((ext_vector_type(4))) unsigned int v4u;
typedef __attribute__((ext_vector_type(8))) int          v8i;
typedef __attribute__((ext_vector_type(4))) int          v4i;

#define NEG_SLOPE 0.2f

// ---------------------------------------------------------------------------
// float atomic max via sign-split int/uint atomics (init memory to -1e30f)
// ---------------------------------------------------------------------------
__device__ __forceinline__ void atomicMaxF32(float* addr, float val) {
  if (val >= 0.0f) atomicMax((int*)addr, __float_as_int(val));
  else             atomicMin((unsigned int*)addr, __float_as_uint(val));
}

__global__ void fill_f32(float* __restrict__ p, float v, long n) {
  long i = (long)blockIdx.x * blockDim.x + threadIdx.x;
  if (i < n) p[i] = v;
}

// ---------------------------------------------------------------------------
// TDM: DMA a row-major 2D fp32 tile [rows x cols] from global into LDS.
// Descriptor layout per CDNA5 ISA 8.3-8.6:
//   group0: count=1 | lds_addr[63:32] | global_addr[120:64] | type=2[127:126]
//   group1: data_size=2(4B)[17:16], tensor_dim0=cols[79:48],
//           tensor_dim1=rows[111:80], tile_dim0=cols[127:112],
//           tile_dim1=rows[143:128], tensor_dim0_stride=cols[207:160],
//           tensor_dim1_stride[255:208]
//   group2/3: higher dims (set to 1 / benign), tile_dim2..4 = 0 (unused)
// This toolchain exposes the 6-arg builtin (clang-23 / therock-10.0 form):
//   (uint32x4, int32x8, int32x4, int32x4, int32x8, i32 cpol)
// ---------------------------------------------------------------------------
__device__ __forceinline__ void tdm_load_2d_to_lds(const void* gsrc, unsigned ldsOff,
                                                   unsigned rows, unsigned cols) {
  unsigned long long ga = (unsigned long long)(uintptr_t)gsrc;
  unsigned long long q;

  v4u g0;
  q = 1ull | ((unsigned long long)ldsOff << 32);          // count=1, lds_addr
  g0[0] = (unsigned)q; g0[1] = (unsigned)(q >> 32);
  q = (ga & 0x01FFFFFFFFFFFFFFull) | (2ull << 62);        // global_addr, type=2
  g0[2] = (unsigned)q; g0[3] = (unsigned)(q >> 32);

  unsigned long long nq = cols, kq = rows;
  v8i g1;
  q = (2ull << 16) | (nq << 48);                          // data_size=4B, dim0 lo16
  g1[0] = (int)q; g1[1] = (int)(q >> 32);
  q = (nq >> 16) | (kq << 16) | (nq << 48);               // dim0 hi, dim1=rows, tile0=cols
  g1[2] = (int)q; g1[3] = (int)(q >> 32);
  q = kq | (nq << 32);                                    // tile1=rows, dim0_stride=cols
  g1[4] = (int)q; g1[5] = (int)(q >> 32);
  q = ((nq * kq) << 16);                                  // dim1_stride = rows*cols
  g1[6] = (int)q; g1[7] = (int)(q >> 32);

  v4i g2; g2[0] = 1; g2[1] = 1; g2[2] = (int)(nq * kq); g2[3] = 0;
  v4i g3; g3[0] = (int)(nq * kq); g3[1] = 1 << 16; g3[2] = 0; g3[3] = 0;
  v8i g4 = {};                                            // unused trailing group

  __builtin_amdgcn_tensor_load_to_lds(g0, g1, g2, g3, g4, 0);
}

// ---------------------------------------------------------------------------
// C[M,NC] = A[M,KK] @ B[KK,NC] (row-major fp32) via V_WMMA_F32_16X16X4_F32.
// B is staged once per workgroup into LDS by the Tensor Data Mover; the
// K-loop then feeds WMMA from ds_load while A streams from global.
// grid.x = M/16 row-tiles, blockDim.x = (NC/16)*32: one wave per 16-col tile.
// A-operand (ISA 7.12.2, 32-bit A 16x4): lanes 0-15 -> M rows, K=k0+{0,1};
// lanes 16-31 -> same rows, K=k0+{2,3}. B mirrors it. C/D: VGPR r, lane l ->
// row (r + 8*(l>>4)), col (l&15).
// ---------------------------------------------------------------------------
template <int KK, int NC>
__global__ void gemm_wmma_f32(const float* __restrict__ A,
                              const float* __restrict__ B,
                              float* __restrict__ C, int M) {
  __shared__ float sB[KK * NC];

  if ((threadIdx.x >> 5) == 0) {
    tdm_load_2d_to_lds(B, (unsigned)(uintptr_t)sB, KK, NC);
    __builtin_amdgcn_s_wait_tensorcnt(0);
  }
  __syncthreads();

  const int lane = threadIdx.x & 31;
  const int nt   = threadIdx.x >> 5;      // which 16-col tile this wave owns
  const int m0   = blockIdx.x * 16;
  const int half = lane >> 4;             // 0: K pair {0,1}, 1: K pair {2,3}
  const int l16  = lane & 15;
  const int n    = nt * 16 + l16;
  const float* arow = A + (size_t)(m0 + l16) * KK;

  v8f c = {};
  for (int k0 = 0; k0 < KK; k0 += 4) {
    const int ka = k0 + 2 * half;         // even -> 8B-aligned float2 load
    v2f a = *(const v2f*)(arow + ka);
    v2f b;
    b.x = sB[ka * NC + n];
    b.y = sB[(ka + 1) * NC + n];
    c = __builtin_amdgcn_wmma_f32_16x16x4_f32(
            /*neg_a=*/false, a, /*neg_b=*/false, b,
            /*c_mod=*/(short)0, c, /*reuse_a=*/false, /*reuse_b=*/false);
  }
#pragma unroll
  for (int r = 0; r < 8; ++r)
    C[(size_t)(m0 + r + 8 * half) * NC + n] = c[r];
}

// ---------------------------------------------------------------------------
// Per-(node,head) attention coefficients: as = <h, a_src>, ad = <h, a_dst>
// ---------------------------------------------------------------------------
__global__ void node_alpha(const float* __restrict__ h,
                           const float* __restrict__ a_src,
                           const float* __restrict__ a_dst,
                           float* __restrict__ as, float* __restrict__ ad,
                           int Nn, int H, int Cc) {
  long t = (long)blockIdx.x * blockDim.x + threadIdx.x;
  if (t >= (long)Nn * H) return;
  int n  = (int)(t / H);
  int hh = (int)(t % H);
  const float* hp = h + (size_t)n * H * Cc + (size_t)hh * Cc;
  const float* sp = a_src + (size_t)hh * Cc;
  const float* dp = a_dst + (size_t)hh * Cc;
  float s = 0.f, d = 0.f;
  for (int c = 0; c < Cc; ++c) { float v = hp[c]; s += v * sp[c]; d += v * dp[c]; }
  as[t] = s; ad[t] = d;
}

// ---------------------------------------------------------------------------
// Edge pass 1: e = LeakyReLU(as[src] + ad[dst]); running max per (dst, head)
// Edge ids >= E0 are the implicit self-loops (src = dst = e - E0).
// ---------------------------------------------------------------------------
__global__ void edge_pass1(const int* __restrict__ srcI, const int* __restrict__ dstI,
                           int E0, int Et, int H,
                           const float* __restrict__ as, const float* __restrict__ ad,
                           float* __restrict__ ebuf, float* __restrict__ emax) {
  long t = (long)blockIdx.x * blockDim.x + threadIdx.x;
  if (t >= (long)Et * H) return;
  int e = (int)(t / H), h = (int)(t % H);
  int s = (e < E0) ? srcI[e] : (e - E0);
  int d = (e < E0) ? dstI[e] : (e - E0);
  float v = as[(size_t)s * H + h] + ad[(size_t)d * H + h];
  v = (v > 0.f) ? v : NEG_SLOPE * v;
  ebuf[t] = v;
  atomicMaxF32(&emax[(size_t)d * H + h], v);
}

// Edge pass 2: p = exp(e - max[dst]); denom[dst] += p
__global__ void edge_pass2(const int* __restrict__ dstI,
                           int E0, int Et, int H,
                           float* __restrict__ ebuf,
                           const float* __restrict__ emax,
                           float* __restrict__ denom) {
  long t = (long)blockIdx.x * blockDim.x + threadIdx.x;
  if (t >= (long)Et * H) return;
  int e = (int)(t / H), h = (int)(t % H);
  int d = (e < E0) ? dstI[e] : (e - E0);
  float p = expf(ebuf[t] - emax[(size_t)d * H + h]);
  ebuf[t] = p;
  atomicAdd(&denom[(size_t)d * H + h], p);
}

// ---------------------------------------------------------------------------
// Edge pass 3 (bandwidth-dominant): (H*C/4) lanes cooperate per edge, one
// float4 (global_load_b128) per lane, scale by alpha, 4 f32 atomics into dst.
// lpeShift = log2(H*C/4): 5 for layer 1 (HC=128), 4 for layer 2 (HC=64).
// ---------------------------------------------------------------------------
__global__ void edge_pass3(const int* __restrict__ srcI, const int* __restrict__ dstI,
                           int E0, int Et, int H, int Cc,
                           int lpeShift, int ccShift,
                           const float* __restrict__ ebuf,
                           const float* __restrict__ denom,
                           const float* __restrict__ feat,
                           float* __restrict__ out) {
  long gt = (long)blockIdx.x * blockDim.x + threadIdx.x;
  int e   = (int)(gt >> lpeShift);
  int sub = (int)(gt & ((1 << lpeShift) - 1));
  if (e >= Et) return;
  int s = (e < E0) ? srcI[e] : (e - E0);
  int d = (e < E0) ? dstI[e] : (e - E0);
  int HC = H * Cc;
  int i  = sub << 2;                      // feature offset of this lane's float4
  int h  = i >> ccShift;                  // head index
  float w = ebuf[(size_t)e * H + h] / denom[(size_t)d * H + h];
  const float4 v = *(const float4*)(feat + (size_t)s * HC + i);
  float* op = out + (size_t)d * HC + i;
  atomicAdd(op + 0, v.x * w);
  atomicAdd(op + 1, v.y * w);
  atomicAdd(op + 2, v.z * w);
  atomicAdd(op + 3, v.w * w);
}

// out[i] = relu(out[i] + bias[i & Fmask])   (F is a power of two: 128 or 64)
__global__ void bias_relu(float* __restrict__ x, const float* __restrict__ b,
                          int Fmask, long n) {
  long i = (long)blockIdx.x * blockDim.x + threadIdx.x;
  if (i >= n) return;
  float v = x[i] + b[i & Fmask];
  x[i] = v > 0.f ? v : 0.f;
}

// Global mean pool: atomic accumulate per graph id
__global__ void pool_accum(const float* __restrict__ h2, const int* __restrict__ batch,
                           float* __restrict__ pooled, float* __restrict__ cnt,
                           int Nn, int Cc) {
  long t = (long)blockIdx.x * blockDim.x + threadIdx.x;
  if (t >= (long)Nn * Cc) return;
  int n = (int)(t / Cc), c = (int)(t % Cc);
  int g = batch[n];
  atomicAdd(&pooled[(size_t)g * Cc + c], h2[t]);
  if (c == 0) atomicAdd(&cnt[g], 1.0f);
}

__global__ void final_linear(const float* __restrict__ pooled,
                             const float* __restrict__ cnt,
                             const float* __restrict__ Wlin,
                             const float* __restrict__ blin,
                             float* __restrict__ out, int Gg, int Cc) {
  int g = blockIdx.x * blockDim.x + threadIdx.x;
  if (g >= Gg) return;
  float inv = 1.0f / fmaxf(cnt[g], 1.0f);
  float acc = 0.f;
  for (int c = 0; c < Cc; ++c) acc += pooled[(size_t)g * Cc + c] * inv * Wlin[c];
  out[g] = acc + blin[0];
}

// ---------------------------------------------------------------------------
static inline int gridFor(long n, int block) { return (int)((n + block - 1) / block); }

extern "C" void kernel_launch(void* const* d_in, const int* in_sizes, int n_in,
                              void* d_out, int out_size, void* d_ws, size_t ws_size,
                              hipStream_t stream) {
  const float* x    = (const float*)d_in[0];
  const int*   ei   = (const int*)  d_in[1];
  const int*   batch= (const int*)  d_in[2];
  const float* W1   = (const float*)d_in[3];
  const float* a1s  = (const float*)d_in[4];
  const float* a1d  = (const float*)d_in[5];
  const float* b1   = (const float*)d_in[6];
  const float* W2   = (const float*)d_in[7];
  const float* a2s  = (const float*)d_in[8];
  const float* a2d  = (const float*)d_in[9];
  const float* b2   = (const float*)d_in[10];
  const float* Wl   = (const float*)d_in[11];
  const float* bl   = (const float*)d_in[12];
  float* out = (float*)d_out;

  const int Nn = in_sizes[0] / 128;   // 100000 nodes (multiple of 16)
  const int E0 = in_sizes[1] / 2;     // 1,600,000 edges
  const int Et = E0 + Nn;             // + self-loops
  const int G  = out_size;            // 64 graphs

  const int* srcI = ei;
  const int* dstI = ei + E0;

  // ---- carve fp32 workspace --------------------------------------------
  float* wp = (float*)d_ws;
  auto carve = [&](long nf) { float* r = wp; wp += nf; return r; };
  float* h1     = carve((long)Nn * 128);  // layer-1 pre-attention features
  float* out1   = carve((long)Nn * 128);  // layer-1 aggregated (then relu'd)
  float* h2     = carve((long)Nn * 64);   // layer-2 pre-attention features
  float* out2   = carve((long)Nn * 64);   // layer-2 aggregated (then relu'd)
  float* as1    = carve((long)Nn * 2);    // alpha_src (reused for layer 2)
  float* ad1    = carve((long)Nn * 2);    // alpha_dst (reused for layer 2)
  float* emax   = carve((long)Nn * 2);
  float* denom  = carve((long)Nn * 2);
  float* ebuf   = carve((long)Et * 2);    // per-edge scores -> exp weights
  float* pooled = carve((long)G * 64);
  float* cnt    = carve((long)G);
  (void)ws_size; (void)n_in;

  const int BLK = 256;

  // ======================= Layer 1 (H=2, C=64) ===========================
  gemm_wmma_f32<128, 128><<<Nn / 16, 8 * 32, 0, stream>>>(x, W1, h1, Nn);
  node_alpha<<<gridFor((long)Nn * 2, BLK), BLK, 0, stream>>>(h1, a1s, a1d, as1, ad1, Nn, 2, 64);

  fill_f32<<<gridFor((long)Nn * 2, BLK), BLK, 0, stream>>>(emax, -1e30f, (long)Nn * 2);
  fill_f32<<<gridFor((long)Nn * 2, BLK), BLK, 0, stream>>>(denom, 0.f, (long)Nn * 2);
  fill_f32<<<gridFor((long)Nn * 128, BLK), BLK, 0, stream>>>(out1, 0.f, (long)Nn * 128);

  edge_pass1<<<gridFor((long)Et * 2, BLK), BLK, 0, stream>>>(srcI, dstI, E0, Et, 2, as1, ad1, ebuf, emax);
  edge_pass2<<<gridFor((long)Et * 2, BLK), BLK, 0, stream>>>(dstI, E0, Et, 2, ebuf, emax, denom);
  edge_pass3<<<gridFor((long)Et << 5, BLK), BLK, 0, stream>>>(srcI, dstI, E0, Et, 2, 64, 5, 6,
                                                              ebuf, denom, h1, out1);
  bias_relu<<<gridFor((long)Nn * 128, BLK), BLK, 0, stream>>>(out1, b1, 127, (long)Nn * 128);

  // ======================= Layer 2 (H=1, C=64) ===========================
  gemm_wmma_f32<128, 64><<<Nn / 16, 4 * 32, 0, stream>>>(out1, W2, h2, Nn);
  node_alpha<<<gridFor((long)Nn, BLK), BLK, 0, stream>>>(h2, a2s, a2d, as1, ad1, Nn, 1, 64);

  fill_f32<<<gridFor((long)Nn, BLK), BLK, 0, stream>>>(emax, -1e30f, (long)Nn);
  fill_f32<<<gridFor((long)Nn, BLK), BLK, 0, stream>>>(denom, 0.f, (long)Nn);
  fill_f32<<<gridFor((long)Nn * 64, BLK), BLK, 0, stream>>>(out2, 0.f, (long)Nn * 64);

  edge_pass1<<<gridFor((long)Et, BLK), BLK, 0, stream>>>(srcI, dstI, E0, Et, 1, as1, ad1, ebuf, emax);
  edge_pass2<<<gridFor((long)Et, BLK), BLK, 0, stream>>>(dstI, E0, Et, 1, ebuf, emax, denom);
  edge_pass3<<<gridFor((long)Et << 4, BLK), BLK, 0, stream>>>(srcI, dstI, E0, Et, 1, 64, 4, 6,
                                                              ebuf, denom, h2, out2);
  bias_relu<<<gridFor((long)Nn * 64, BLK), BLK, 0, stream>>>(out2, b2, 63, (long)Nn * 64);

  // ================= Global mean pool + final linear =====================
  fill_f32<<<gridFor((long)G * 64, BLK), BLK, 0, stream>>>(pooled, 0.f, (long)G * 64);
  fill_f32<<<1, 64, 0, stream>>>(cnt, 0.f, (long)G);
  pool_accum<<<gridFor((long)Nn * 64, BLK), BLK, 0, stream>>>(out2, batch, pooled, cnt, Nn, 64);
  final_linear<<<1, 64, 0, stream>>>(pooled, cnt, Wl, bl, out, G, 64);
}